// compress_jpeg_5128190951921
// MI455X (gfx1250) — compile-verified
//
#include <hip/hip_runtime.h>

typedef __attribute__((ext_vector_type(2))) float v2f;
typedef __attribute__((ext_vector_type(4))) float v4f;
typedef __attribute__((ext_vector_type(8))) float v8f;

// cos(k*pi/16)
#define C1 0.98078528040323044913f
#define C2 0.92387953251128675613f
#define C3 0.83146961230254523708f
#define C4 0.70710678118654752440f
#define C5 0.55557023301960222474f
#define C6 0.38268343236508977173f
#define C7 0.19509032201612826785f

// C8[x][u] = cos((2x+1)*u*pi/16), row-major
__constant__ float c8tab[64] = {
    1.f,  C1,  C2,  C3,  C4,  C5,  C6,  C7,
    1.f,  C3,  C6, -C7, -C4, -C1, -C2, -C5,
    1.f,  C5, -C6, -C1, -C4,  C7,  C2,  C3,
    1.f,  C7, -C2, -C5,  C4,  C3, -C6, -C1,
    1.f, -C7, -C2,  C5,  C4, -C3, -C6,  C1,
    1.f, -C5, -C6,  C1, -C4, -C7,  C2, -C3,
    1.f, -C3,  C6,  C7, -C4,  C1, -C2,  C5,
    1.f, -C1,  C2, -C3,  C4, -C5,  C6, -C7};

__constant__ float c_qy[64] = {
    16,11,10,16,24,40,51,61,
    12,12,14,19,26,58,60,55,
    14,13,16,24,40,57,69,56,
    14,17,22,29,51,87,80,62,
    18,22,37,56,68,109,103,77,
    24,35,55,64,81,104,113,92,
    49,64,78,87,103,121,120,101,
    72,92,95,98,112,100,103,99};

__constant__ float c_qc[64] = {
    17,18,24,47,99,99,99,99,
    18,21,26,66,99,99,99,99,
    24,26,56,99,99,99,99,99,
    47,66,99,99,99,99,99,99,
    99,99,99,99,99,99,99,99,
    99,99,99,99,99,99,99,99,
    99,99,99,99,99,99,99,99,
    99,99,99,99,99,99,99,99};

// One block = one 64x64 pixel region of one image.
//   phase 1: 256 threads color-convert + 2x2-pool into LDS (float4 traffic)
//   phase 2: 8 waves x 3 tiles each (16 Y tiles + 4 Cb + 4 Cr), each tile DCT =
//            D2^T * X * D2 with D2 = blkdiag(C8,C8) via 2x(4 chained
//            v_wmma_f32_16x16x4_f32), then scale/quant/round-to-even, store.
__global__ __launch_bounds__(256) void jpeg_fused_kernel(
    const float* __restrict__ img,  // [32,3,512,512]
    float* __restrict__ yq,         // [32,4096,8,8]
    float* __restrict__ cbq,        // [32,1024,8,8]
    float* __restrict__ crq)        // [32,1024,8,8]
{
    __shared__ float yl[64 * 80];        // Y-128, stride 80 (conflict-free halves)
    __shared__ float cbl[32 * 40];       // pooled Cb (128-shift cancels), stride 40
    __shared__ float crl[32 * 40];       // pooled Cr
    __shared__ float pbuf[8 * 16 * 20];  // per-wave P staging, stride 20

    const int t    = threadIdx.x;
    const int lane = t & 31;
    const int wv   = t >> 5;
    const int n    = lane & 15;   // column within 16x16 tile
    const int half = lane >> 4;   // 0: lanes 0-15, 1: lanes 16-31

    // region decode: 2048 = 32 images * 8*8 regions of 64x64 px
    const int b  = blockIdx.x >> 6;
    const int rr = blockIdx.x & 63;
    const int rh = rr >> 3;
    const int rw = rr & 7;

    // ---------------- phase 1: color convert + chroma pooling ----------------
    {
        const float* rp = img + (size_t)b * 3 * 512 * 512;
        const float* gp = rp + 512 * 512;
        const float* bp = gp + 512 * 512;
        const int py0 = rh * 64;
        const int px0 = rw * 64;
#pragma unroll
        for (int s2 = 0; s2 < 2; ++s2) {
            const int sidx = t + 256 * s2;       // strip: 4 wide x 2 tall
            const int srow = sidx >> 4;          // 0..31
            const int scol = sidx & 15;          // 0..15
            const size_t o0 = (size_t)(py0 + srow * 2) * 512 + px0 + scol * 4;
            v4f r0 = *(const v4f*)(rp + o0), r1 = *(const v4f*)(rp + o0 + 512);
            v4f g0 = *(const v4f*)(gp + o0), g1 = *(const v4f*)(gp + o0 + 512);
            v4f b0 = *(const v4f*)(bp + o0), b1 = *(const v4f*)(bp + o0 + 512);

            v4f y0  = 255.f * (0.299f * r0 + 0.587f * g0 + 0.114f * b0) - 128.f;
            v4f y1  = 255.f * (0.299f * r1 + 0.587f * g1 + 0.114f * b1) - 128.f;
            v4f cb0 = 255.f * (-0.168736f * r0 - 0.331264f * g0 + 0.5f * b0);
            v4f cb1 = 255.f * (-0.168736f * r1 - 0.331264f * g1 + 0.5f * b1);
            v4f cr0 = 255.f * (0.5f * r0 - 0.418688f * g0 - 0.081312f * b0);
            v4f cr1 = 255.f * (0.5f * r1 - 0.418688f * g1 - 0.081312f * b1);

            *(v4f*)&yl[(srow * 2 + 0) * 80 + scol * 4] = y0;
            *(v4f*)&yl[(srow * 2 + 1) * 80 + scol * 4] = y1;
            cbl[srow * 40 + scol * 2 + 0] = 0.25f * (cb0.x + cb0.y + cb1.x + cb1.y);
            cbl[srow * 40 + scol * 2 + 1] = 0.25f * (cb0.z + cb0.w + cb1.z + cb1.w);
            crl[srow * 40 + scol * 2 + 0] = 0.25f * (cr0.x + cr0.y + cr1.x + cr1.y);
            crl[srow * 40 + scol * 2 + 1] = 0.25f * (cr0.z + cr0.w + cr1.z + cr1.w);
        }
    }

    // ---------------- per-lane init (amortized over 3 tiles) -----------------
    // D2 fragment: same scalars act as A-layout of D2^T (mul1) and B-layout of
    // D2 (mul2). Zero-block condition reduces to (j<2) != (n<8).
    float d2f[4][2];
#pragma unroll
    for (int j = 0; j < 4; ++j)
#pragma unroll
        for (int i = 0; i < 2; ++i) {
            const int k = 4 * j + 2 * half + i;  // row of D2
            const float cv = c8tab[(k & 7) * 8 + (n & 7)];
            d2f[j][i] = ((j < 2) == (n < 8)) ? cv : 0.f;
        }

    // fused scale DCT_SCALE[u][v]/Q[u][v]; v fixed per lane, u = accumulator row
    const int vc = lane & 7;
    const float av = (vc == 0) ? C4 : 1.f;
    float qsY[8], qsC[8];
#pragma unroll
    for (int r = 0; r < 8; ++r) {
        const float num = 0.25f * ((r == 0) ? C4 : 1.f) * av;
        qsY[r] = num / c_qy[r * 8 + vc];
        qsC[r] = num / c_qc[r * 8 + vc];
    }

    __syncthreads();

    // ---------------- phase 2: 3 tiles per wave ------------------------------
    float* pb = &pbuf[wv * 16 * 20];
    for (int s = 0; s < 3; ++s) {
        const int tile = wv + 8 * s;             // 0..15 Y, 16..19 Cb, 20..23 Cr
        const int chroma = tile >= 16;
        const float* src; int sstride, soff;
        float* outp; int th, tw, bps; size_t nimg;
        if (!chroma) {
            const int tr = tile >> 2, tc = tile & 3;
            src = yl; sstride = 80; soff = tr * 16 * 80 + tc * 16;
            outp = yq; th = rh * 4 + tr; tw = rw * 4 + tc; bps = 64; nimg = 4096;
        } else {
            const int ci = tile - 16;
            const int iscr = ci >= 4;
            const int tr = (ci & 3) >> 1, tc = ci & 1;
            src = iscr ? crl : cbl; sstride = 40; soff = tr * 16 * 40 + tc * 16;
            outp = iscr ? crq : cbq;
            th = rh * 2 + tr; tw = rw * 2 + tc; bps = 32; nimg = 1024;
        }

        // multiply 1: P = D2^T * X  (4 chained K=4 f32 WMMAs)
        v8f P = {0.f, 0.f, 0.f, 0.f, 0.f, 0.f, 0.f, 0.f};
#pragma unroll
        for (int j = 0; j < 4; ++j) {
            v2f a = {d2f[j][0], d2f[j][1]};
            const int row0 = 4 * j + 2 * half;
            v2f bb = {src[soff + (row0 + 0) * sstride + n],
                      src[soff + (row0 + 1) * sstride + n]};
            P = __builtin_amdgcn_wmma_f32_16x16x4_f32(false, a, false, bb,
                                                      (short)0, P, false, false);
        }

        // C/D layout -> A layout via per-wave LDS staging (same-wave = in order)
#pragma unroll
        for (int r = 0; r < 8; ++r)
            pb[(r + 8 * half) * 20 + n] = P[r];
        float pf[4][2];
#pragma unroll
        for (int j = 0; j < 4; ++j)
#pragma unroll
            for (int i = 0; i < 2; ++i)
                pf[j][i] = pb[n * 20 + 4 * j + 2 * half + i];

        // multiply 2: Out = P * D2
        v8f O = {0.f, 0.f, 0.f, 0.f, 0.f, 0.f, 0.f, 0.f};
#pragma unroll
        for (int j = 0; j < 4; ++j) {
            v2f a = {pf[j][0], pf[j][1]};
            v2f bb = {d2f[j][0], d2f[j][1]};
            O = __builtin_amdgcn_wmma_f32_16x16x4_f32(false, a, false, bb,
                                                      (short)0, O, false, false);
        }

        // quantize, round (RNE = jnp.round), store in block-split layout
        const int dr = half, dc = n >> 3;
        const int nblk = (th * 2 + dr) * bps + (tw * 2 + dc);
        float* ob = outp + ((size_t)b * nimg + (size_t)nblk) * 64 + vc;
#pragma unroll
        for (int r = 0; r < 8; ++r) {
            const float sc = chroma ? qsC[r] : qsY[r];
            ob[r * 8] = rintf(O[r] * sc);
        }
    }
}

extern "C" void kernel_launch(void* const* d_in, const int* in_sizes, int n_in,
                              void* d_out, int out_size, void* d_ws, size_t ws_size,
                              hipStream_t stream) {
    (void)in_sizes; (void)n_in; (void)out_size; (void)d_ws; (void)ws_size;
    const float* img = (const float*)d_in[0];
    float* yq  = (float*)d_out;                       // 32*4096*64
    float* cbq = yq  + (size_t)32 * 4096 * 64;        // 32*1024*64
    float* crq = cbq + (size_t)32 * 1024 * 64;        // 32*1024*64
    // 32 images * 64 regions of 64x64 px = 2048 blocks, 256 threads (8 waves)
    jpeg_fused_kernel<<<2048, 256, 0, stream>>>(img, yq, cbq, crq);
}